// Clique2NodeConvBasic_3547642987231
// MI455X (gfx1250) — compile-verified
//
#include <hip/hip_runtime.h>
#include <hip/hip_bf16.h>

typedef __attribute__((ext_vector_type(2))) float v2f;
typedef __attribute__((ext_vector_type(8))) float v8f;

#define D_FEAT 128

// -------------------------------------------------------------------------
// Kernel 1: edge-parallel scatter-sum.  One wave32 per edge: each lane loads
// a float4 (16B) of the 512B clique row and issues 4 native f32 atomics into
// the node accumulator.  x_clique (25.6MB) and the accumulator (51.2MB) are
// L2-resident on MI455X (192MB L2), so this is an L2-atomic-throughput phase.
// -------------------------------------------------------------------------
__global__ void c2n_scatter_sum_kernel(const float* __restrict__ x_clique,
                                       const long long* __restrict__ node_idx,
                                       const long long* __restrict__ clique_idx,
                                       float* __restrict__ sum,
                                       float* __restrict__ cnt,
                                       int n_edges) {
    const int waves_per_block = blockDim.x >> 5;
    const int e = blockIdx.x * waves_per_block + (threadIdx.x >> 5);
    if (e >= n_edges) return;
    const int lane = threadIdx.x & 31;

    const long long c = clique_idx[e];
    const long long n = node_idx[e];

    // 32 lanes x 16B = 512B = one full feature row
    const float4 v = ((const float4*)(x_clique + (size_t)c * D_FEAT))[lane];
    float* dst = sum + (size_t)n * D_FEAT + lane * 4;
    unsafeAtomicAdd(dst + 0, v.x);
    unsafeAtomicAdd(dst + 1, v.y);
    unsafeAtomicAdd(dst + 2, v.z);
    unsafeAtomicAdd(dst + 3, v.w);
    if (lane == 0) unsafeAtomicAdd(cnt + n, 1.0f);
}

// -------------------------------------------------------------------------
// Kernel 2: fused scatter-mean + Linear via V_WMMA_F32_16X16X4_F32.
//   out[m,n] = (sum[m,:]/max(cnt[m],1)) . W[n,:] + b[n]
// One 256-thread block (8 waves) per 16-row tile; wave w owns output columns
// [16w, 16w+16).  K=128 consumed in 32 WMMA steps of K=4.
//
// Fragment layouts per CDNA5 ISA (f32, wave32):
//   A 16x4 : lanes 0-15  -> M=lane,   v0=K(k0),   v1=K(k0+1)
//            lanes 16-31 -> M=lane-16,v0=K(k0+2), v1=K(k0+3)
//   B 4x16 : lanes 0-15  -> N=lane,   v0=K(k0),   v1=K(k0+1)
//            lanes 16-31 -> N=lane-16,v0=K(k0+2), v1=K(k0+3)
//   C/D 16x16: VGPR i -> M=i (lanes 0-15) / M=i+8 (lanes 16-31), N=lane&15
// -------------------------------------------------------------------------
__global__ void c2n_mean_gemm_wmma_kernel(const float* __restrict__ sum,
                                          const float* __restrict__ cnt,
                                          const float* __restrict__ W,
                                          const float* __restrict__ bias,
                                          float* __restrict__ out,
                                          int n_nodes) {
    const int lane    = threadIdx.x & 31;
    const int wave    = threadIdx.x >> 5;
    const int hi      = lane >> 4;          // 0: lanes 0-15, 1: lanes 16-31
    const int lane16  = lane & 15;
    const int rowbase = blockIdx.x * 16;
    const int row     = rowbase + lane16;   // A-matrix row for this lane
    const int ncol    = wave * 16 + lane16; // B/D column for this lane

    // scatter-mean scale, folded into the A fragment
    float rinv = 0.0f;
    int arow = row;
    if (row < n_nodes) {
        rinv = 1.0f / fmaxf(cnt[row], 1.0f);
    } else {
        arow = n_nodes - 1;   // safe address; contribution zeroed by rinv=0
    }

    const float* Arow = sum + (size_t)arow * D_FEAT + 2 * hi;
    const float* Brow = W   + (size_t)ncol * D_FEAT + 2 * hi;

    v8f acc = {};
#pragma unroll
    for (int k0 = 0; k0 < D_FEAT; k0 += 4) {
        const float2 av = *(const float2*)(Arow + k0);
        const float2 bv = *(const float2*)(Brow + k0);
        v2f a, b;
        a.x = av.x * rinv;
        a.y = av.y * rinv;
        b.x = bv.x;
        b.y = bv.y;
        // 8 args: (neg_a, A, neg_b, B, c_mod, C, reuse_a, reuse_b)
        acc = __builtin_amdgcn_wmma_f32_16x16x4_f32(
            false, a, false, b, (short)0, acc, false, false);
    }

    const float bb = bias[ncol];
#pragma unroll
    for (int i = 0; i < 8; ++i) {
        const int m = rowbase + i + 8 * hi;
        if (m < n_nodes) {
            out[(size_t)m * D_FEAT + ncol] = acc[i] + bb;
        }
    }
}

// -------------------------------------------------------------------------
// Launch.  Inputs (setup_inputs order):
//   d_in[0] x              [N_NODES,128]  f32  (unused by the reference math)
//   d_in[1] x_clique       [N_CLIQUES,128] f32
//   d_in[2] node2clique    [2, E]          int64 (row0=node_idx, row1=clique_idx)
//   d_in[3] W              [128,128]       f32
//   d_in[4] b              [128]           f32
// Output: [N_NODES,128] f32.
// Workspace: sum [N_NODES*128] f32 followed by cnt [N_NODES] f32.
// -------------------------------------------------------------------------
extern "C" void kernel_launch(void* const* d_in, const int* in_sizes, int n_in,
                              void* d_out, int out_size, void* d_ws, size_t ws_size,
                              hipStream_t stream) {
    const float*     x_clique = (const float*)d_in[1];
    const long long* n2c      = (const long long*)d_in[2];
    const float*     W        = (const float*)d_in[3];
    const float*     b        = (const float*)d_in[4];
    float*           out      = (float*)d_out;

    const int n_nodes = in_sizes[0] / D_FEAT;
    const int n_edges = in_sizes[2] / 2;

    float* sum = (float*)d_ws;
    float* cnt = sum + (size_t)n_nodes * D_FEAT;
    const size_t zero_bytes = ((size_t)n_nodes * D_FEAT + (size_t)n_nodes) * sizeof(float);
    hipMemsetAsync(d_ws, 0, zero_bytes, stream);

    // Scatter phase: 8 waves / block, 1 edge / wave.
    const int waves_per_block = 8;
    const int blocks1 = (n_edges + waves_per_block - 1) / waves_per_block;
    c2n_scatter_sum_kernel<<<blocks1, 256, 0, stream>>>(
        x_clique, n2c /*node_idx*/, n2c + n_edges /*clique_idx*/, sum, cnt, n_edges);

    // Fused mean + Linear (WMMA f32): one block per 16 output rows.
    const int blocks2 = (n_nodes + 15) / 16;
    c2n_mean_gemm_wmma_kernel<<<blocks2, 256, 0, stream>>>(
        sum, cnt, W, b, out, n_nodes);
}